// SeparateTSQLinearNodeBlock_84198538871460
// MI455X (gfx1250) — compile-verified
//
#include <hip/hip_runtime.h>

typedef float v2f __attribute__((ext_vector_type(2)));
typedef float v8f __attribute__((ext_vector_type(8)));

#define MUL0 32
#define MUL1 16
#define DIM  80
#define XSTR 84                                // padded LDS node stride (bank-conflict-free)
#define WF00_SZ (1024*32)                      // 32768 floats
#define WF11_SZ (256*32)                       //  8192 floats
#define WF01_SZ (512*16)                       //  8192 floats
#define WF_TOTAL (WF00_SZ + WF11_SZ + WF01_SZ) // 49152 floats = 192 KB
#define NTHREADS 512
#define WAVES  16
#define TILE_F (16*DIM)                        // 1280 floats per 16-node tile

__device__ __forceinline__ v8f wmma4(v2f a, v2f b, v8f c) {
  // D(16x16,f32) = A(16x4,f32) x B(4x16,f32) + C  -- native f32 WMMA on gfx1250
  return __builtin_amdgcn_wmma_f32_16x16x4_f32(false, a, false, b, (short)0, c,
                                               false, false);
}
__device__ __forceinline__ v2f lds2(const float* p) { return *(const v2f*)p; }

// ---------------------------------------------------------------------------
// Prep: fold Linear (L0/L1) + normalizations into weights AND swizzle into the
// WMMA-B fragment-native layout:
//   logical Wf[k][n]  ->  off = kb*(NT*64) + tile*64 + nn*4 + h*2 + parity
//   with kb=k>>2, h=(k&3)>>1, parity=k&1, tile=n>>4, nn=n&15, NT=#16-col tiles.
// A lane then fetches its B fragment {row 4kb+2h+0, row 4kb+2h+1} with ONE
// aligned ds_load_b64 at base + nn*16B + h*8B (contiguous, conflict-free).
// ---------------------------------------------------------------------------
__global__ void fold_weights_kernel(const float* __restrict__ W00,
                                    const float* __restrict__ W01,
                                    const float* __restrict__ W11,
                                    const float* __restrict__ L0,
                                    const float* __restrict__ L1,
                                    float* __restrict__ wf) {
  const float S00 = 0.00552427172802f;   // 1/(32*sqrt(32))
  const float S11 = 0.00637886681534f;   // 1/(sqrt(3)*16*sqrt(32))
  const float S01 = 0.01104854345604f;   // 1/(sqrt(512)*sqrt(16))
  int idx = blockIdx.x * blockDim.x + threadIdx.x;
  if (idx < WF00_SZ) {
    int k = idx >> 5, vp = idx & 31;
    float s = 0.f;
#pragma unroll
    for (int w = 0; w < 32; ++w) s += W00[k*32 + w] * L0[w*32 + vp];
    int off = (k >> 2)*128 + (vp >> 4)*64 + (vp & 15)*4 + ((k & 3) >> 1)*2 + (k & 1);
    wf[off] = s * S00;
  } else if (idx < WF00_SZ + WF11_SZ) {
    int j = idx - WF00_SZ; int k = j >> 5, vp = j & 31;
    float s = 0.f;
#pragma unroll
    for (int w = 0; w < 32; ++w) s += W11[k*32 + w] * L0[w*32 + vp];
    int off = (k >> 2)*128 + (vp >> 4)*64 + (vp & 15)*4 + ((k & 3) >> 1)*2 + (k & 1);
    wf[WF00_SZ + off] = s * S11;
  } else if (idx < WF_TOTAL) {
    int j = idx - (WF00_SZ + WF11_SZ); int k = j >> 4, vp = j & 15;
    float s = 0.f;
#pragma unroll
    for (int w = 0; w < 16; ++w) s += W01[k*16 + w] * L1[w*16 + vp];
    int off = (k >> 2)*64 + vp*4 + ((k & 3) >> 1)*2 + (k & 1);
    wf[WF00_SZ + WF11_SZ + off] = s * S01;
  }
}

// ---------------------------------------------------------------------------
// Main pass: one wave owns a 16-node tile; 16 waves/block -> 4 waves/SIMD.
// Per u-iteration: issue ALL fragment loads (DS), then ALL WMMAs, so the
// scheduler gets prefetch distance (partial s_wait_dscnt instead of 0-waits).
// ---------------------------------------------------------------------------
__global__ __launch_bounds__(NTHREADS, 1)
void tsq_pass_kernel(const float* __restrict__ x, const float* __restrict__ wf,
                     float* __restrict__ out, int ntiles, int accumulate) {
  extern __shared__ float smem[];
  float* sW00 = smem;                       // swizzled [256 kb][2 tile][16 n][2 h][2]
  float* sW11 = smem + WF00_SZ;             // swizzled [64 kb][2][16][2][2]
  float* sW01 = smem + WF00_SZ + WF11_SZ;   // swizzled [128 kb][16][2][2]
  float* sX   = smem + WF_TOTAL;            // [WAVES][16][XSTR]

  {
    const float4* s4 = (const float4*)wf;
    float4* d4 = (float4*)smem;
    for (int j = threadIdx.x; j < WF_TOTAL/4; j += blockDim.x) d4[j] = s4[j];
  }
  __syncthreads();

  const int lane = threadIdx.x & 31;
  const int wave = threadIdx.x >> 5;
  const int m    = lane & 15;      // node-in-tile / B-fragment column
  const int half = lane >> 4;      // K-half of A/B fragments, row-half of C
  const int koff = half * 2;
  const int mh   = m*4 + half*2;   // B-fragment lane offset within 64-float block

  float* sXw = sX + wave * (16*XSTR);
  const float* Xp = sXw + m * XSTR;          // this lane's node (80 f32, padded)

  const long tstride = (long)gridDim.x * WAVES;
  for (long t = (long)blockIdx.x * WAVES + wave; t < ntiles; t += tstride) {
    // Stage 16 nodes x 80 f32 -> LDS with padded stride 84.
    {
      const float4* s4 = (const float4*)(x + t*(long)TILE_F + m*DIM + half*40);
      float4* d4 = (float4*)(sXw + m*XSTR + half*40);
#pragma unroll
      for (int j = 0; j < 10; ++j) d4[j] = s4[j];
    }
    asm volatile("s_wait_dscnt 0" ::: "memory"); // wave-local LDS visibility

    v8f acc0a = {}; // out cols  0..15 (scalar channels)
    v8f acc0b = {}; // out cols 16..31
    v8f acc10 = {}; // l=1 channels, i = 0
    v8f acc11 = {}; // i = 1
    v8f acc12 = {}; // i = 2

    // ---- path 0e x 0e -> 0e : K = 1024 = (u,v) -----------------------------
    for (int u = 0; u < MUL0; ++u) {
      const float xu = Xp[u];
      const float* wb = sW00 + u*1024;       // 8 kb-blocks of 128 floats
      v2f a[8], b0[8], b1[8];
#pragma unroll
      for (int s = 0; s < 8; ++s) {
        a[s]  = lds2(Xp + s*4 + koff) * xu;
        b0[s] = lds2(wb + s*128      + mh);
        b1[s] = lds2(wb + s*128 + 64 + mh);
      }
#pragma unroll
      for (int s = 0; s < 8; ++s) {
        acc0a = wmma4(a[s], b0[s], acc0a);
        acc0b = wmma4(a[s], b1[s], acc0b);
      }
    }

    // ---- path 1o x 1o -> 0e : K = 256 = (u,v), A = <x1_u, x1_v> ------------
    for (int u = 0; u < MUL1; ++u) {
      const float* xu = Xp + MUL0 + u*3;
      const float xu0 = xu[0], xu1 = xu[1], xu2 = xu[2];
      const float* wb = sW11 + u*512;        // 4 kb-blocks of 128 floats
      v2f a[4], b0[4], b1[4];
#pragma unroll
      for (int s = 0; s < 4; ++s) {
        const float* xv = Xp + MUL0 + (s*4 + koff)*3;
        a[s].x = xu0*xv[0] + xu1*xv[1] + xu2*xv[2];
        a[s].y = xu0*xv[3] + xu1*xv[4] + xu2*xv[5];
        b0[s] = lds2(wb + s*128      + mh);
        b1[s] = lds2(wb + s*128 + 64 + mh);
      }
#pragma unroll
      for (int s = 0; s < 4; ++s) {
        acc0a = wmma4(a[s], b0[s], acc0a);
        acc0b = wmma4(a[s], b1[s], acc0b);
      }
    }

    // ---- path 0e x 1o -> 1o : K = 512 = (u,v), per Cartesian i; B shared ---
    for (int u = 0; u < MUL0; ++u) {
      const float xu = Xp[u];
      const float* wb = sW01 + u*256;        // 4 kb-blocks of 64 floats
      v2f a0[4], a1[4], a2[4], b[4];
#pragma unroll
      for (int s = 0; s < 4; ++s) {
        const float* xv = Xp + MUL0 + (s*4 + koff)*3;
        a0[s].x = xu*xv[0]; a0[s].y = xu*xv[3];
        a1[s].x = xu*xv[1]; a1[s].y = xu*xv[4];
        a2[s].x = xu*xv[2]; a2[s].y = xu*xv[5];
        b[s] = lds2(wb + s*64 + mh);
      }
#pragma unroll
      for (int s = 0; s < 4; ++s) {
        acc10 = wmma4(a0[s], b[s], acc10);
        acc11 = wmma4(a1[s], b[s], acc11);
        acc12 = wmma4(a2[s], b[s], acc12);
      }
    }

    // ---- write back (C layout: vgpr r -> row r + 8*half, col = lane&15) ----
    float* obase = out + (t * 16 + half * 8) * (long)DIM;
    if (accumulate) {
#pragma unroll
      for (int r = 0; r < 8; ++r) {
        float* p = obase + r * DIM;
        p[m]            += acc0a[r];
        p[16 + m]       += acc0b[r];
        p[32 + 3*m]     += acc10[r];
        p[32 + 3*m + 1] += acc11[r];
        p[32 + 3*m + 2] += acc12[r];
      }
    } else {
#pragma unroll
      for (int r = 0; r < 8; ++r) {
        float* p = obase + r * DIM;
        p[m]             = acc0a[r];
        p[16 + m]        = acc0b[r];
        p[32 + 3*m]      = acc10[r];
        p[32 + 3*m + 1]  = acc11[r];
        p[32 + 3*m + 2]  = acc12[r];
      }
    }
  }
}

extern "C" void kernel_launch(void* const* d_in, const int* in_sizes, int n_in,
                              void* d_out, int out_size, void* d_ws, size_t ws_size,
                              hipStream_t stream) {
  (void)n_in; (void)out_size; (void)ws_size;
  const float* feats = (const float*)d_in[0];
  const float* msgs  = (const float*)d_in[1];
  const float* W00_1 = (const float*)d_in[2];
  const float* W01_1 = (const float*)d_in[3];
  const float* W11_1 = (const float*)d_in[4];
  const float* W00_2 = (const float*)d_in[5];
  const float* W01_2 = (const float*)d_in[6];
  const float* W11_2 = (const float*)d_in[7];
  const float* L0    = (const float*)d_in[8];
  const float* L1    = (const float*)d_in[9];
  float* out = (float*)d_out;
  float* wf  = (float*)d_ws;   // [2][WF_TOTAL] folded+swizzled weights (384 KB)

  const int n = in_sizes[0] / DIM;
  const int ntiles = n / 16;

  const int fold_blocks = (WF_TOTAL + 255) / 256;
  fold_weights_kernel<<<fold_blocks, 256, 0, stream>>>(W00_1, W01_1, W11_1, L0, L1, wf);
  fold_weights_kernel<<<fold_blocks, 256, 0, stream>>>(W00_2, W01_2, W11_2, L0, L1, wf + WF_TOTAL);

  const size_t shmem = (size_t)(WF_TOTAL + WAVES * 16 * XSTR) * sizeof(float); // 276 KB
  (void)hipFuncSetAttribute(reinterpret_cast<const void*>(tsq_pass_kernel),
                            hipFuncAttributeMaxDynamicSharedMemorySize, (int)shmem);
  const int grid = 512; // 8192 waves, grid-stride over 62500 tiles
  tsq_pass_kernel<<<grid, NTHREADS, shmem, stream>>>(feats, wf,            out, ntiles, 0);
  tsq_pass_kernel<<<grid, NTHREADS, shmem, stream>>>(msgs,  wf + WF_TOTAL, out, ntiles, 1);
}